// EdgeScorePredictor_13511967113600
// MI455X (gfx1250) — compile-verified
//
#include <hip/hip_runtime.h>

// ---------------------------------------------------------------------------
// EdgeScorePredictor, restructured:
//   P[n, 0:16]  = h[n] . W[:, 0:128]^T      (src projection)
//   P[n, 16:32] = h[n] . W[:, 128:256]^T    (dst projection)
//   out[e, c]   = P[src[e], c] + P[dst[e], 16+c] + b[c]
// Kernel 1: dense GEMM via V_WMMA_F32_16X16X4_F32 (native f32 matrix op).
// Kernel 2: L2-resident gather + add (P is 12.8 MB, fits easily in 192MB L2).
// ---------------------------------------------------------------------------

typedef __attribute__((ext_vector_type(2))) float v2f;
typedef __attribute__((ext_vector_type(8))) float v8f;

#define N_NODES 100000
#define N_EDGES 600000
#define DIM     128
#define NCH     16

// One wave per 16-node tile. Computes both 16-wide channel halves.
__global__ __launch_bounds__(256) void node_proj_wmma(
    const float* __restrict__ h,    // (N_NODES, 128)
    const float* __restrict__ W,    // (16, 256) row-major
    float* __restrict__ P)          // (N_NODES, 32)
{
    const int lane  = threadIdx.x & 31;
    const int wave  = (blockIdx.x * (blockDim.x >> 5)) + (threadIdx.x >> 5);
    const int m0    = wave * 16;
    if (m0 >= N_NODES) return;           // wave-uniform: EXEC stays all-ones

    const int half  = lane >> 4;         // 0: lanes 0-15, 1: lanes 16-31
    const int sub   = lane & 15;
    const int khalf = half * 2;          // K sub-offset per WMMA layout

    // A fragment source: row m0+sub of h, K pairs {k+khalf, k+khalf+1}
    const float* arow = h + (size_t)(m0 + sub) * DIM + khalf;
    // B fragment source: channel `sub`; W row-major so K is contiguous.
    //   tile0 (src half): W[sub][k]        tile1 (dst half): W[sub][128+k]
    const float* brow0 = W + (size_t)sub * (2 * DIM) + khalf;
    const float* brow1 = brow0 + DIM;

    v8f acc0 = {};   // channels 0..15  (src projection)
    v8f acc1 = {};   // channels 16..31 (dst projection)

#pragma unroll
    for (int k = 0; k < DIM; k += 4) {
        v2f a  = *(const v2f*)(arow  + k);
        v2f b0 = *(const v2f*)(brow0 + k);
        v2f b1 = *(const v2f*)(brow1 + k);
        acc0 = __builtin_amdgcn_wmma_f32_16x16x4_f32(
            false, a, false, b0, (short)0, acc0, false, false);
        acc1 = __builtin_amdgcn_wmma_f32_16x16x4_f32(
            false, a, false, b1, (short)0, acc1, false, false);
    }

    // C/D layout: VGPR r -> row m0 + r + 8*half, col = sub (+16 for tile1)
    const int mrw = m0 + half * 8;
#pragma unroll
    for (int r = 0; r < 8; ++r) {
        float* prow = P + (size_t)(mrw + r) * 32;
        prow[sub]      = acc0[r];
        prow[16 + sub] = acc1[r];
    }
}

// One thread per (edge, 4 channels): float4 gathers out of L2-resident P.
__global__ __launch_bounds__(256) void edge_gather_add(
    const float* __restrict__ P,       // (N_NODES, 32)
    const int*   __restrict__ esrc,    // (N_EDGES,)
    const int*   __restrict__ edst,    // (N_EDGES,)
    const float* __restrict__ b,       // (16,)
    float*       __restrict__ out)     // (N_EDGES, 16)
{
    const int t = blockIdx.x * blockDim.x + threadIdx.x;
    const int e  = t >> 2;
    if (e >= N_EDGES) return;
    const int c4 = (t & 3) * 4;

    const int s = esrc[e];
    const int d = edst[e];

    const float4 ps = *(const float4*)(P + (size_t)s * 32 + c4);
    const float4 pd = *(const float4*)(P + (size_t)d * 32 + 16 + c4);
    const float4 bb = *(const float4*)(b + c4);

    float4 r;
    r.x = ps.x + pd.x + bb.x;
    r.y = ps.y + pd.y + bb.y;
    r.z = ps.z + pd.z + bb.z;
    r.w = ps.w + pd.w + bb.w;

    *(float4*)(out + (size_t)e * NCH + c4) = r;
}

extern "C" void kernel_launch(void* const* d_in, const int* in_sizes, int n_in,
                              void* d_out, int out_size, void* d_ws, size_t ws_size,
                              hipStream_t stream) {
    // setup_inputs() order: h, edge_src, edge_dst, W, b
    const float* h    = (const float*)d_in[0];
    const int*   src  = (const int*)  d_in[1];
    const int*   dst  = (const int*)  d_in[2];
    const float* W    = (const float*)d_in[3];
    const float* b    = (const float*)d_in[4];
    float*       out  = (float*)d_out;
    float*       P    = (float*)d_ws;   // needs N_NODES*32*4 = 12.8 MB

    // Kernel 1: 6250 M-tiles, 8 waves (256 threads) per block.
    const int waves_needed = (N_NODES + 15) / 16;                 // 6250
    const int blocks1 = (waves_needed + 7) / 8;                   // 782
    node_proj_wmma<<<blocks1, 256, 0, stream>>>(h, W, P);

    // Kernel 2: one thread per (edge, 4ch) -> 2.4M threads.
    const int threads2 = N_EDGES * 4;
    const int blocks2 = (threads2 + 255) / 256;                   // 9375
    edge_gather_add<<<blocks2, 256, 0, stream>>>(P, src, dst, b, out);
}